// Dominant_26860725469614
// MI455X (gfx1250) — compile-verified
//
#include <hip/hip_runtime.h>

typedef __attribute__((ext_vector_type(2))) float v2f;
typedef __attribute__((ext_vector_type(8))) float v8f;

#define NN   12288
#define FEATD 512
#define HIDD  128

// ---------------- degree / normalization ----------------
__global__ void k_init_deg(float* deg, int n) {
  int i = blockIdx.x * blockDim.x + threadIdx.x;
  if (i < n) deg[i] = 1.0f;            // self-loop weight folded in
}

__global__ void k_accum_deg(float* deg, const int* __restrict__ dst,
                            const float* __restrict__ w, int E) {
  int e = blockIdx.x * blockDim.x + threadIdx.x;
  if (e < E) atomicAdd(&deg[dst[e]], w[e]);
}

__global__ void k_dinv(const float* __restrict__ deg, float* dinv, int n) {
  int i = blockIdx.x * blockDim.x + threadIdx.x;
  if (i < n) { float d = deg[i]; dinv[i] = d > 0.f ? rsqrtf(d) : 0.f; }
}

__global__ void k_norm(const float* __restrict__ dinv, const int* __restrict__ src,
                       const int* __restrict__ dst, const float* __restrict__ w,
                       float* norm, int E) {
  int e = blockIdx.x * blockDim.x + threadIdx.x;
  if (e < E) norm[e] = dinv[src[e]] * w[e] * dinv[dst[e]];
}

__global__ void k_zero(float* p, unsigned n) {
  unsigned i = blockIdx.x * blockDim.x + threadIdx.x;
  if (i < n) p[i] = 0.f;
}

// ---------------- fp32 WMMA GEMM ----------------
// C[M,Nc] = A[M,K] (row major) * B.
//   TRANSB=false: B row-major [K,Nc]      (weight matrices)
//   TRANSB=true : B element (k,n) = B[n*K + k]  (s @ s^T; C stored nontemporal)
// One wave computes a 32(M) x 64(N) tile; 8 waves/block stack along M (256 rows/block).
// M % 256 == 0, Nc % 64 == 0, K % 4 == 0 at all call sites -> EXEC all-1s everywhere.
template <bool TRANSB>
__global__ void __launch_bounds__(256)
k_gemm_wmma(const float* __restrict__ A, const float* __restrict__ B,
            float* __restrict__ C, int M, int Nc, int K) {
  const int lane = threadIdx.x & 31;
  const int wave = threadIdx.x >> 5;
  const int half = lane >> 4;      // K-half select (ISA 16x4 A layout)
  const int l15  = lane & 15;      // M (for A) / N (for B,C)
  const int m0 = (blockIdx.y * 8 + wave) * 32;
  const int nb = blockIdx.x * 64;

  const v8f vz = {0.f,0.f,0.f,0.f,0.f,0.f,0.f,0.f};
  v8f acc[2][4];
#pragma unroll
  for (int t = 0; t < 2; ++t)
#pragma unroll
    for (int j = 0; j < 4; ++j) acc[t][j] = vz;

  // fold the lane-half K offset into the base pointers once; per-iteration
  // addresses are then base + k0 (constant immediates after unroll)
  const float* a0p = A + (size_t)(m0 + l15) * K + half * 2;
  const float* a1p = a0p + (size_t)16 * K;

  const float* bp[4];
#pragma unroll
  for (int j = 0; j < 4; ++j) {
    const int col = nb + j * 16 + l15;
    if (TRANSB) bp[j] = B + (size_t)col * K + half * 2;   // adjacent pair -> b64 load
    else        bp[j] = B + (size_t)(half * 2) * Nc + col;
  }

  for (int k0 = 0; k0 < K; k0 += 4) {
    // issue the whole fragment load clause first, keep all fragments live so
    // the compiler staggers s_wait_loadcnt across the 8-WMMA burst
    v2f a0 = *reinterpret_cast<const v2f*>(a0p + k0);
    v2f a1 = *reinterpret_cast<const v2f*>(a1p + k0);
    v2f bb[4];
#pragma unroll
    for (int j = 0; j < 4; ++j) {
      if (TRANSB) {
        bb[j] = *reinterpret_cast<const v2f*>(bp[j] + k0);
      } else {
        const float* p = bp[j] + (size_t)k0 * Nc;
        bb[j].x = p[0];
        bb[j].y = p[Nc];
      }
    }
#pragma unroll
    for (int j = 0; j < 4; ++j) {
      acc[0][j] = __builtin_amdgcn_wmma_f32_16x16x4_f32(
          false, a0, false, bb[j], (short)0, acc[0][j], false, false);
      acc[1][j] = __builtin_amdgcn_wmma_f32_16x16x4_f32(
          false, a1, false, bb[j], (short)0, acc[1][j], false, false);
    }
  }

  // D layout: VGPR v, lanes 0-15 -> M=v ; lanes 16-31 -> M=v+8
#pragma unroll
  for (int t = 0; t < 2; ++t)
#pragma unroll
    for (int j = 0; j < 4; ++j)
#pragma unroll
      for (int v = 0; v < 8; ++v) {
        const int row = m0 + t * 16 + half * 8 + v;
        const int col = nb + j * 16 + l15;
        float* cp = C + (size_t)row * Nc + col;
        if (TRANSB) __builtin_nontemporal_store(acc[t][j][v], cp);  // 604MB stream
        else        *cp = acc[t][j][v];
      }
}

// ---------------- edge scatter: agg[dst] += norm * h[src] ----------------
// one thread per (edge, 4-float chunk); F is a power of two; shift = log2(F/4)
__global__ void k_scatter(const float* __restrict__ h, float* __restrict__ agg,
                          const int* __restrict__ src, const int* __restrict__ dst,
                          const float* __restrict__ norm, int E, int F, int shift) {
  unsigned idx = blockIdx.x * blockDim.x + threadIdx.x;
  unsigned total = (unsigned)E << shift;
  if (idx >= total) return;
  unsigned e = idx >> shift;
  unsigned c = idx & ((1u << shift) - 1u);
  int s = src[e], d = dst[e];
  float w = norm[e];
  const float4 v = *((const float4*)(h + (size_t)s * F) + c);
  float* ap = agg + (size_t)d * F + (size_t)c * 4;
  atomicAdd(ap + 0, w * v.x);
  atomicAdd(ap + 1, w * v.y);
  atomicAdd(ap + 2, w * v.z);
  atomicAdd(ap + 3, w * v.w);
}

// ---------------- out = relu(agg + dinv[i]^2 * h + bias) ----------------
__global__ void k_finalize(const float* __restrict__ agg, const float* __restrict__ h,
                           const float* __restrict__ dinv, const float* __restrict__ bias,
                           float* __restrict__ out, unsigned total, int fshift) {
  unsigned idx = blockIdx.x * blockDim.x + threadIdx.x;
  if (idx >= total) return;
  unsigned i = idx >> fshift;
  unsigned f = idx & ((1u << fshift) - 1u);
  float di = dinv[i];
  float v = agg[idx] + di * di * h[idx] + bias[f];
  out[idx] = v > 0.f ? v : 0.f;
}

// ---------------- host side ----------------
static inline unsigned ceil_div(unsigned a, unsigned b) { return (a + b - 1) / b; }

static void run_conv(const float* in, int K, const float* W, const float* bias, int F,
                     float* h, float* agg, float* out,
                     const int* src, const int* dst, const float* norm,
                     const float* dinv, int E, hipStream_t stream) {
  dim3 gb((unsigned)F / 64u, (unsigned)NN / 256u);
  k_gemm_wmma<false><<<gb, 256, 0, stream>>>(in, W, h, NN, F, K);
  unsigned nf = (unsigned)NN * (unsigned)F;
  k_zero<<<ceil_div(nf, 256u), 256, 0, stream>>>(agg, nf);
  int shift = __builtin_ctz((unsigned)F) - 2;           // log2(F/4)
  unsigned tot = (unsigned)E << shift;
  k_scatter<<<ceil_div(tot, 256u), 256, 0, stream>>>(h, agg, src, dst, norm, E, F, shift);
  int fshift = __builtin_ctz((unsigned)F);              // log2(F)
  k_finalize<<<ceil_div(nf, 256u), 256, 0, stream>>>(agg, h, dinv, bias, out, nf, fshift);
}

extern "C" void kernel_launch(void* const* d_in, const int* in_sizes, int n_in,
                              void* d_out, int out_size, void* d_ws, size_t ws_size,
                              hipStream_t stream) {
  const float* x   = (const float*)d_in[0];
  const int*   ei  = (const int*)d_in[1];
  const float* ew  = (const float*)d_in[2];
  const float* W1  = (const float*)d_in[3];  const float* b1  = (const float*)d_in[4];
  const float* W2  = (const float*)d_in[5];  const float* b2  = (const float*)d_in[6];
  const float* Wa1 = (const float*)d_in[7];  const float* ba1 = (const float*)d_in[8];
  const float* Wa2 = (const float*)d_in[9];  const float* ba2 = (const float*)d_in[10];
  const float* Ws1 = (const float*)d_in[11]; const float* bs1 = (const float*)d_in[12];

  const int E = in_sizes[2];                 // edge_weight count
  const int* src = ei;
  const int* dst = ei + E;

  // workspace layout (fp32)
  float* ws   = (float*)d_ws;
  float* deg  = ws;                          // NN
  float* dinv = deg  + NN;                   // NN
  float* norm = dinv + NN;                   // E
  float* buf1 = norm + E;                    // NN*HIDD  (gemm out h)
  float* buf2 = buf1 + (size_t)NN * HIDD;    // NN*HIDD  (agg)
  float* buf3 = buf2 + (size_t)NN * HIDD;    // NN*HIDD  (h1 / xh / s)
  float* buf4 = buf3 + (size_t)NN * HIDD;    // NN*HIDD  (h2)
  float* big1 = buf4 + (size_t)NN * HIDD;    // NN*FEATD (h for attr decoder 2)
  float* big2 = big1 + (size_t)NN * FEATD;   // NN*FEATD (agg for attr decoder 2)

  float* out   = (float*)d_out;
  float* a_hat = out;                               // NN*NN
  float* x_hat = out + (size_t)NN * NN;             // NN*FEATD

  // gcn_norm: deg = 1 (self loop) + sum_w per dst; norm = dinv[src]*w*dinv[dst]
  k_init_deg<<<ceil_div(NN, 256u), 256, 0, stream>>>(deg, NN);
  k_accum_deg<<<ceil_div((unsigned)E, 256u), 256, 0, stream>>>(deg, dst, ew, E);
  k_dinv<<<ceil_div(NN, 256u), 256, 0, stream>>>(deg, dinv, NN);
  k_norm<<<ceil_div((unsigned)E, 256u), 256, 0, stream>>>(dinv, src, dst, ew, norm, E);

  // shared encoder
  run_conv(x,    FEATD, W1,  b1,  HIDD, buf1, buf2, buf3, src, dst, norm, dinv, E, stream); // h1
  run_conv(buf3, HIDD,  W2,  b2,  HIDD, buf1, buf2, buf4, src, dst, norm, dinv, E, stream); // h2
  // attribute decoder
  run_conv(buf4, HIDD,  Wa1, ba1, HIDD, buf1, buf2, buf3, src, dst, norm, dinv, E, stream); // xh
  run_conv(buf3, HIDD,  Wa2, ba2, FEATD, big1, big2, x_hat, src, dst, norm, dinv, E, stream); // x_hat
  // structure decoder
  run_conv(buf4, HIDD,  Ws1, bs1, HIDD, buf1, buf2, buf3, src, dst, norm, dinv, E, stream); // s

  // a_hat = s @ s^T  (transposed-B view of s; nontemporal C stream)
  dim3 ga((unsigned)NN / 64u, (unsigned)NN / 256u);
  k_gemm_wmma<true><<<ga, 256, 0, stream>>>(buf3, buf3, a_hat, NN, NN, HIDD);
}